// MMATopKLayer_77618648973843
// MI455X (gfx1250) — compile-verified
//
#include <hip/hip_runtime.h>
#include <stdint.h>

// ---------------------------------------------------------------------------
// MI455X top-K (K=512) lexicographic selection, one workgroup per batch.
//  * per-wave async-to-LDS staging (global_load_async_to_lds_b128) overlapped
//    with per-wave histogramming -> single HBM read, ~5.6us roofline
//  * 12-bit radix histogram on ordered-x; shuffle-based block scan (2 barriers)
//  * gather <=4096 candidates as packed 64-bit (x,y) keys
//  * bitonic sort with barrier-free intra-wave (j<16) passes on wave32
// ---------------------------------------------------------------------------

typedef int v4i __attribute__((ext_vector_type(4)));
typedef __attribute__((address_space(1))) v4i* g4p_t;   // global int4*
typedef __attribute__((address_space(3))) v4i* l4p_t;   // LDS int4*

static __device__ __forceinline__ uint32_t f2ord(float f) {
    uint32_t u = __float_as_uint(f);
    return (u & 0x80000000u) ? ~u : (u | 0x80000000u);
}
static __device__ __forceinline__ float ord2f(uint32_t o) {
    uint32_t u = (o & 0x80000000u) ? (o & 0x7fffffffu) : ~o;
    return __uint_as_float(u);
}

constexpr int KTOP  = 512;
constexpr int HBINS = 4096;   // top 12 bits of ordered x
constexpr int CAP   = 4096;   // candidate buffer (power of two for bitonic)
constexpr int NTHR  = 1024;   // threads per block (32 waves, wave32)

extern "C" __global__ void __launch_bounds__(NTHR, 1)
mma_topk_kernel(const float* __restrict__ corners,
                const int*   __restrict__ lengths,
                float*       __restrict__ out,
                int M)
{
    extern __shared__ char smem[];
    const int b    = blockIdx.x;
    const int tid  = threadIdx.x;
    const int bd   = blockDim.x;
    const int lane = tid & 31;
    const int wave = tid >> 5;
    const int nwaves = bd >> 5;

    float*              data  = (float*)smem;                                   // 8*M bytes
    unsigned*           hist  = (unsigned*)(smem + (size_t)M * 8);              // 16 KB
    unsigned long long* cand  = (unsigned long long*)(smem + (size_t)M * 8 + HBINS * 4); // 32 KB
    unsigned*           scanb = (unsigned*)(smem + (size_t)M * 8 + HBINS * 4 + (size_t)CAP * 8); // wave totals
    unsigned*           ctrl  = scanb + 64;                                     // control words

    int n = lengths[b];
    n = n < 0 ? 0 : (n > M ? M : n);
    const int Ksel = n < KTOP ? n : KTOP;

    // ---- zero histogram + counters first (before any LDS atomics) ----
    for (int i = tid; i < HBINS; i += bd) hist[i] = 0u;
    if (tid == 0) { ctrl[0] = 0u; ctrl[1] = 0u; }
    __syncthreads();

    // ---- per-wave pipelined staging + histogram ----
    // wave w owns a contiguous slice; it waits only on its OWN async copies
    // before histogramming, so copy and compute overlap across the 32 waves.
    const float* src  = corners + (size_t)b * M * 2;
    const int    nvec = (M * 2) / 4;            // 16-byte chunks total
    const int    vpw  = nvec / nwaves;          // chunks per wave
    const int    rpw  = M / nwaves;             // rows per wave
    const int    vbeg = wave * vpw;
    const int    vend = (wave == nwaves - 1) ? nvec : vbeg + vpw;

#if __has_builtin(__builtin_amdgcn_global_load_async_to_lds_b128)
    for (int i = vbeg + lane; i < vend; i += 32) {
        __builtin_amdgcn_global_load_async_to_lds_b128(
            (g4p_t)(void*)(src + 4 * i), (l4p_t)(void*)(data + 4 * i), 0, 0);
    }
#if __has_builtin(__builtin_amdgcn_s_wait_asynccnt)
    __builtin_amdgcn_s_wait_asynccnt(0);
#else
    asm volatile("s_wait_asynccnt 0" ::: "memory");
#endif
#else
    {   // fallback: vector loads through VGPRs
        const float4* s4 = (const float4*)src;
        float4*       d4 = (float4*)data;
        for (int i = vbeg + lane; i < vend; i += 32) d4[i] = s4[i];
    }
#endif

    // histogram of top 12 bits of ordered x over this wave's valid rows
    {
        const int rbeg = wave * rpw;
        const int rend = (wave == nwaves - 1) ? n : (rbeg + rpw < n ? rbeg + rpw : n);
        for (int i = rbeg + lane; i < rend; i += 32) {
            uint32_t xo = f2ord(data[2 * i]);
            atomicAdd(&hist[xo >> 20], 1u);
        }
    }
    __syncthreads();   // all waves waited on own copies -> all LDS data visible

    // ---- threshold search: seg-sums + shuffle scan (2 barriers total) ----
    {
        const int SEG = HBINS / NTHR;  // 4 bins per thread
        unsigned hloc[SEG];
        unsigned s = 0;
        for (int q = 0; q < SEG; ++q) { hloc[q] = hist[SEG * tid + q]; s += hloc[q]; }

        // intra-wave inclusive scan of s (wave32 shuffles, no barriers)
        unsigned incl_w = s;
        for (int d = 1; d < 32; d <<= 1) {
            unsigned t = __shfl_up(incl_w, d, 32);
            if (lane >= d) incl_w += t;
        }
        if (lane == 31) scanb[wave] = incl_w;   // wave totals
        __syncthreads();
        if (wave == 0) {                        // scan the 32 wave totals
            unsigned wt = scanb[lane];
            unsigned iw = wt;
            for (int d = 1; d < 32; d <<= 1) {
                unsigned t = __shfl_up(iw, d, 32);
                if (lane >= d) iw += t;
            }
            scanb[lane] = iw - wt;              // exclusive wave offset
        }
        __syncthreads();

        unsigned incl = incl_w + scanb[wave];
        unsigned excl = incl - s;
        if (excl < (unsigned)Ksel && (unsigned)Ksel <= incl) {  // unique thread
            unsigned c = excl, T = (unsigned)(SEG * tid);
            for (int q = 0; q < SEG; ++q) {
                c += hloc[q];
                if (c >= (unsigned)Ksel) { T = (unsigned)(SEG * tid + q); break; }
            }
            ctrl[0] = T;
        }
    }
    __syncthreads();
    const unsigned T = ctrl[0];

    // ---- gather candidates (bin <= T) as packed 64-bit keys ----
    for (int i = tid; i < n; i += bd) {
        uint32_t xo = f2ord(data[2 * i]);
        if ((xo >> 20) <= T) {
            unsigned pos = atomicAdd(&ctrl[1], 1u);
            if (pos < (unsigned)CAP) {
                uint32_t yo = f2ord(data[2 * i + 1]);
                cand[pos] = ((unsigned long long)xo << 32) | (unsigned long long)yo;
            }
        }
    }
    __syncthreads();
    unsigned c1 = ctrl[1];
    if (c1 > (unsigned)CAP) c1 = (unsigned)CAP;

    // ---- dynamically-sized bitonic sort (sortN in {512,...,4096}) ----
    unsigned sortN = 512;
    while (sortN < c1) sortN <<= 1;
    for (int i = (int)c1 + tid; i < (int)sortN; i += bd) cand[i] = ~0ull;
    __syncthreads();   // padding visible before sort

    // wave32 insight: for j <= 16, element e and partner e^j are inside one
    // 32-aligned block owned by a single wave (threads tid and tid^j share
    // tid>>5), and per-wave LDS ops complete in order -> no block barrier.
    // Barrier only for cross-wave passes (j >= 32) and at j == 16 to make
    // the preceding cross-wave pass visible.
    for (unsigned k = 2; k <= sortN; k <<= 1) {
        for (unsigned j = k >> 1; j >= 1; j >>= 1) {
            if (j >= 16) __syncthreads();
            for (unsigned i = (unsigned)tid; i < sortN; i += (unsigned)bd) {
                unsigned p = i ^ j;
                if (p > i) {
                    bool up = ((i & k) == 0u);
                    unsigned long long a = cand[i], c = cand[p];
                    if ((a > c) == up) { cand[i] = c; cand[p] = a; }
                }
            }
        }
    }
    __syncthreads();

    // ---- decode + write [2K] floats, zero padding beyond Ksel ----
    float* ob = out + (size_t)b * (2 * KTOP);
    for (int j = tid; j < KTOP; j += bd) {
        float xv = 0.0f, yv = 0.0f;
        if (j < Ksel) {
            unsigned long long kk = cand[j];
            xv = ord2f((uint32_t)(kk >> 32));
            yv = ord2f((uint32_t)(kk & 0xffffffffull));
        }
        ob[2 * j]     = xv;
        ob[2 * j + 1] = yv;
    }
}

extern "C" void kernel_launch(void* const* d_in, const int* in_sizes, int n_in,
                              void* d_out, int out_size, void* d_ws, size_t ws_size,
                              hipStream_t stream) {
    const float* corners = (const float*)d_in[0];
    const int*   lengths = (const int*)d_in[1];
    float*       out     = (float*)d_out;

    const int B = in_sizes[1];
    const int M = in_sizes[0] / (2 * B);

    const size_t smem_bytes = (size_t)M * 8            // staged batch data
                            + (size_t)HBINS * 4        // histogram
                            + (size_t)CAP * 8          // candidate keys
                            + 64 * 4                   // wave-total scan buffer
                            + 16;                      // control words
    mma_topk_kernel<<<B, NTHR, smem_bytes, stream>>>(corners, lengths, out, M);
}